// SmallModel_17471926960776
// MI455X (gfx1250) — compile-verified
//
#include <hip/hip_runtime.h>
#include <hip/hip_bf16.h>

// ---------------------------------------------------------------------------
// The 62-layer chain is purely affine (no activations), so it collapses to
//   out = x @ Mtot.T + ctot   with Mtot: [4,16], ctot: [4].
// Kernel 1 (tiny, 1 block) folds the chain in f64 into d_ws.
// Kernel 2 streams x (268MB) -> out (67MB) at HBM rate using f32 WMMA.
// ---------------------------------------------------------------------------

typedef float v2f __attribute__((ext_vector_type(2)));
typedef float v8f __attribute__((ext_vector_type(8)));

// ---------------------------------------------------------------------------
// Kernel 1: collapse W2 * (W60 * ... * W1) and biases, in double precision.
// 128 threads: thread (r, col) owns element M[r][col] of the evolving 8x16 map.
// ---------------------------------------------------------------------------
__global__ void collapse_kernel(const float* __restrict__ W1,
                                const float* __restrict__ b1,
                                const float* __restrict__ Ws,
                                const float* __restrict__ bs,
                                const float* __restrict__ W2,
                                const float* __restrict__ b2,
                                float* __restrict__ Mc /* 64 + 4 floats */) {
    __shared__ double M[2][8][16];
    __shared__ double c[2][8];

    const int tid = threadIdx.x;      // 128 threads
    const int r   = tid >> 4;         // 0..7
    const int col = tid & 15;         // 0..15

    // M0 = W1 (maps 16 -> 8), c0 = b1
    M[0][r][col] = (double)W1[r * 16 + col];
    if (col == 0) c[0][r] = (double)b1[r];
    __syncthreads();

    int cur = 0;
    for (int i = 0; i < 60; ++i) {
        const float* W = Ws + i * 64;   // 8x8 layer i
        double acc = 0.0;
        #pragma unroll
        for (int k = 0; k < 8; ++k)
            acc += (double)W[r * 8 + k] * M[cur][k][col];
        M[cur ^ 1][r][col] = acc;
        if (col == 0) {
            double cacc = (double)bs[i * 8 + r];
            #pragma unroll
            for (int k = 0; k < 8; ++k)
                cacc += (double)W[r * 8 + k] * c[cur][k];
            c[cur ^ 1][r] = cacc;
        }
        __syncthreads();
        cur ^= 1;
    }

    // Final: Mtot = W2 (4x8) @ M (8x16); ctot = W2 @ c + b2
    if (tid < 64) {
        const int r2 = tid >> 4;        // 0..3
        double acc = 0.0;
        #pragma unroll
        for (int k = 0; k < 8; ++k)
            acc += (double)W2[r2 * 8 + k] * M[cur][k][col];
        Mc[r2 * 16 + col] = (float)acc; // row-major [4][16]
        if (col == 0) {
            double cacc = (double)b2[r2];
            #pragma unroll
            for (int k = 0; k < 8; ++k)
                cacc += (double)W2[r2 * 8 + k] * c[cur][k];
            Mc[64 + r2] = (float)cacc;
        }
    }
}

// ---------------------------------------------------------------------------
// Kernel 2: out[B,4] = x[B,16] @ Mtot.T + ctot via V_WMMA_F32_16X16X4_F32.
// Each wave32 owns 16 batch rows (the WMMA M dimension); K=16 handled with
// 4 chained WMMAs; N columns 0..3 are the 4 outputs (4..15 multiply zeros).
// A layout (16x4 f32): lane = m + 16*hi holds K = 2*hi + {0,1} in v0,v1.
// B layout (4x16 f32): lane = n + 16*hi holds K = 2*hi + {0,1} in v0,v1.
// C/D layout: VGPR j, lane = n + 16*hi -> element D[j + 8*hi][n].
// ---------------------------------------------------------------------------
__global__ void mlp_wmma_kernel(const float* __restrict__ x,
                                const float* __restrict__ Mc,
                                float* __restrict__ out,
                                int num_tiles) {
    const int lane = threadIdx.x & 31;
    const int col  = lane & 15;   // N index (output feature for col<4)
    const int hi   = lane >> 4;   // K sub-pair selector
    const int wave  = (int)((blockIdx.x * blockDim.x + threadIdx.x) >> 5);
    const int nwave = (int)((gridDim.x * blockDim.x) >> 5);

    // Build B slices: B_s[k][n] = Mtot[n][4s + k], zero for n >= 4.
    v2f bmat[4];
    #pragma unroll
    for (int s = 0; s < 4; ++s) {
        if (col < 4) {
            const float* p = Mc + col * 16 + s * 4 + hi * 2;
            bmat[s][0] = p[0];
            bmat[s][1] = p[1];
        } else {
            bmat[s][0] = 0.0f;
            bmat[s][1] = 0.0f;
        }
    }
    // Bias broadcast through the first WMMA's C operand (same value per column).
    const float bias = (col < 4) ? Mc[64 + col] : 0.0f;
    v8f cinit;
    #pragma unroll
    for (int j = 0; j < 8; ++j) cinit[j] = bias;

    for (int t = wave; t < num_tiles; t += nwave) {
        const size_t row0 = (size_t)t * 16;
        // A-slices: this lane covers row (row0+col), K = 4s + 2*hi + {0,1}.
        const float* px = x + (row0 + (size_t)col) * 16 + hi * 2;
        v2f a0 = *(const v2f*)(px + 0);
        v2f a1 = *(const v2f*)(px + 4);
        v2f a2 = *(const v2f*)(px + 8);
        v2f a3 = *(const v2f*)(px + 12);

        v8f acc;
        acc = __builtin_amdgcn_wmma_f32_16x16x4_f32(false, a0, false, bmat[0],
                                                    (short)0, cinit, false, false);
        acc = __builtin_amdgcn_wmma_f32_16x16x4_f32(false, a1, false, bmat[1],
                                                    (short)0, acc, false, false);
        acc = __builtin_amdgcn_wmma_f32_16x16x4_f32(false, a2, false, bmat[2],
                                                    (short)0, acc, false, false);
        acc = __builtin_amdgcn_wmma_f32_16x16x4_f32(false, a3, false, bmat[3],
                                                    (short)0, acc, false, false);

        if (col < 4) {
            // D[j + 8*hi][col] -> out[(row0 + j + 8*hi)*4 + col]
            float* po = out + (row0 + (size_t)hi * 8) * 4 + col;
            #pragma unroll
            for (int j = 0; j < 8; ++j) po[j * 4] = acc[j];
        }
    }
}

// Tail: rows not covered by full 16-row tiles (B is 4M here, so normally none).
__global__ void mlp_tail_kernel(const float* __restrict__ x,
                                const float* __restrict__ Mc,
                                float* __restrict__ out,
                                int row_start, int nrows) {
    int i = blockIdx.x * blockDim.x + threadIdx.x;
    if (i >= nrows) return;
    size_t r = (size_t)row_start + i;
    const float* px = x + r * 16;
    #pragma unroll
    for (int o = 0; o < 4; ++o) {
        float acc = Mc[64 + o];
        #pragma unroll
        for (int k = 0; k < 16; ++k) acc += px[k] * Mc[o * 16 + k];
        out[r * 4 + o] = acc;
    }
}

extern "C" void kernel_launch(void* const* d_in, const int* in_sizes, int n_in,
                              void* d_out, int out_size, void* d_ws, size_t ws_size,
                              hipStream_t stream) {
    const float* x  = (const float*)d_in[0];
    const float* W1 = (const float*)d_in[1];
    const float* b1 = (const float*)d_in[2];
    const float* Ws = (const float*)d_in[3];
    const float* bs = (const float*)d_in[4];
    const float* W2 = (const float*)d_in[5];
    const float* b2 = (const float*)d_in[6];
    float* out = (float*)d_out;
    float* Mc  = (float*)d_ws;   // 64 floats Mtot + 4 floats ctot

    // 1) Fold the affine chain (tiny).
    hipLaunchKernelGGL(collapse_kernel, dim3(1), dim3(128), 0, stream,
                       W1, b1, Ws, bs, W2, b2, Mc);

    // 2) Bandwidth-bound WMMA apply.
    const int B     = in_sizes[0] / 16;   // batch rows
    const int tiles = B / 16;
    if (tiles > 0) {
        hipLaunchKernelGGL(mlp_wmma_kernel, dim3(2048), dim3(256), 0, stream,
                           x, Mc, out, tiles);
    }
    const int rem = B - tiles * 16;
    if (rem > 0) {
        hipLaunchKernelGGL(mlp_tail_kernel, dim3((rem + 255) / 256), dim3(256),
                           0, stream, x, Mc, out, tiles * 16, rem);
    }
}